// LSTMModel_23819888623782
// MI455X (gfx1250) — compile-verified
//
#include <hip/hip_runtime.h>

// ---------------------------------------------------------------------------
// Bidirectional 5-layer LSTM (H=5) on MI455X / gfx1250.
//
//  * Each 64-thread block (2 wave32s) owns a 16-row batch tile. Wave 0 runs
//    the forward direction, wave 1 the backward direction of each layer.
//    Batch elements are independent -> no inter-block sync.
//  * Per-step gate GEMM uses V_WMMA_F32_16X16X4_F32 with gates on M and
//    batch on N: A = weight tile (VGPR-resident per layer/dir),
//    B = [x;h]^T K-chunk. One accumulator per gate type (i,f,g,o) so the
//    cell update is purely elementwise in the WMMA D layout.
//  * K packing: x features at k in [0,DIN), h feedback at k in [DIN,DIN+5),
//    padded to a multiple of 4 -> 4 chunks (16 WMMAs/step) for layers >=1,
//    2 chunks (8 WMMAs/step) for layer 0. This minimizes the chained-WMMA
//    depth on the latency-critical recurrence.
//  * Layer activations ping-pong through two [T=512][B=2048][10] f32
//    buffers in d_ws (2 x 41.9 MB = 83.9 MB, resident in the 192MB L2).
//  * x-chunk loads for step t+1 are issued before the step-t WMMAs so L2
//    latency hides behind the matrix ops.
// ---------------------------------------------------------------------------

typedef __attribute__((ext_vector_type(2))) float v2f;
typedef __attribute__((ext_vector_type(8))) float v8f;

#define Hdim 5
#define NB   2048
#define NT   512
#define BTILE 16

__device__ __forceinline__ float fast_sig(float x) {
  // 1 / (1 + exp(-x)) via v_exp_f32 + v_rcp_f32
  return __builtin_amdgcn_rcpf(1.0f + __builtin_amdgcn_exp2f(-x * 1.4426950408889634f));
}
__device__ __forceinline__ float fast_tanh(float x) {
  // tanh(x) = 2*sigmoid(2x) - 1
  return 2.0f * __builtin_amdgcn_rcpf(1.0f + __builtin_amdgcn_exp2f(-x * 2.8853900817779268f)) - 1.0f;
}

__device__ __forceinline__ float lane_lo_bcast(float v, int lane) {
  // lanes 16..31 fetch the value held by lane (L-16); lanes 0..15 keep theirs
  int i = __builtin_amdgcn_ds_bpermute((lane & 15) << 2, __builtin_bit_cast(int, v));
  return __builtin_bit_cast(float, i);
}

__device__ __forceinline__ v8f wmma4(v2f a, v2f b, v8f c) {
  // D(16x16,f32) = A(16x4,f32) x B(4x16,f32) + C
  return __builtin_amdgcn_wmma_f32_16x16x4_f32(
      /*neg_a=*/false, a, /*neg_b=*/false, b,
      /*c_mod=*/(short)0, c, /*reuse_a=*/false, /*reuse_b=*/false);
}

// K layout: k in [0, DIN)        -> Wi columns (x features)
//           k in [DIN, DIN+5)    -> Wh columns (h feedback)
//           k in [DIN+5, 4*NCH)  -> zero pad
template <int DIN>
__device__ __forceinline__ void lstm_layer_dir(
    const float* __restrict__ src, float* __restrict__ dst,
    const float* __restrict__ Wi, const float* __restrict__ Wh,
    const float* __restrict__ bias, int b0, int lane, int dir) {
  constexpr int NCH = (DIN + Hdim + 3) / 4;  // 2 for DIN=2, 4 for DIN=10
  constexpr int CH2 = NCH - 2;               // chunk with {x...|h0,h1}
  constexpr int CH3 = NCH - 1;               // chunk with {h2,h3|h4,0}
  const bool fwd  = (dir == 0);
  const int  r    = lane & 15;               // A: M row / B,D: N column
  const int  koff = (lane >= 16) ? 2 : 0;    // K offset for the high half-wave

  // ---- Preload weights in WMMA A layout: a[gate_type][chunk] -------------
  v2f a[4][NCH];
#pragma unroll
  for (int g = 0; g < 4; ++g) {
#pragma unroll
    for (int c = 0; c < NCH; ++c) {
#pragma unroll
      for (int j = 0; j < 2; ++j) {
        int k = 4 * c + koff + j;
        float v = 0.0f;
        if (r < Hdim) {
          int gate = g * Hdim + r;
          if (k < DIN)                          v = Wi[gate * DIN + k];
          else if (k < DIN + Hdim)              v = Wh[gate * Hdim + (k - DIN)];
        }
        a[g][c][j] = v;
      }
    }
  }
  // Bias as the C initializer (element e = hidden unit e of this gate type)
  v8f bc[4];
#pragma unroll
  for (int g = 0; g < 4; ++g) {
#pragma unroll
    for (int e = 0; e < 8; ++e) bc[g][e] = (e < Hdim) ? bias[g * Hdim + e] : 0.0f;
  }

  // ---- x-chunk loader (B layout: lane = batch col, half-wave = K+2) ------
  const int bl = b0 + r;
  auto loadx = [&](int t, v2f (&xb)[NCH]) {
    if constexpr (DIN == 2) {            // layer 0: x is [B][T][2]
      const float* p = src + ((size_t)bl * NT + t) * 2;
      v2f v = {0.0f, 0.0f};
      if (lane < 16) v = *(const v2f*)p; // chunk0 lo = {x0,x1}; hi = h (merged later)
      xb[0] = v;
      xb[1] = (v2f){0.0f, 0.0f};
    } else {                             // layers >=1: buf is [T][B][10]
      const float* p = src + ((size_t)t * NB + bl) * 10;
      xb[0] = *(const v2f*)(p + koff);        // K 0..3
      xb[1] = *(const v2f*)(p + 4 + koff);    // K 4..7
      v2f v = {0.0f, 0.0f};
      if (lane < 16) v = *(const v2f*)(p + 8); // chunk2 lo = {x8,x9}; hi = h
      xb[2] = v;
      xb[3] = (v2f){0.0f, 0.0f};
    }
  };

  float hc[Hdim], cc[Hdim];
#pragma unroll
  for (int e = 0; e < Hdim; ++e) { hc[e] = 0.0f; cc[e] = 0.0f; }

  v2f xc[NCH], xn[NCH];
  loadx(fwd ? 0 : NT - 1, xc);

  for (int s = 0; s < NT; ++s) {
    const int t = fwd ? s : (NT - 1 - s);
    // Prefetch next step's x (independent of the recurrence)
    if (s + 1 < NT) loadx(fwd ? t + 1 : t - 1, xn);
    else {
#pragma unroll
      for (int c = 0; c < NCH; ++c) xn[c] = (v2f){0.0f, 0.0f};
    }

    // Merge h feedback into the last two B-chunks:
    //   chunk CH2: lo = x tail (or x0,x1 for layer 0), hi = {h0,h1}
    //   chunk CH3: lo = {h2,h3},                       hi = {h4,0}
    const bool lo = lane < 16;
    float h0b = lane_lo_bcast(hc[0], lane);
    float h1b = lane_lo_bcast(hc[1], lane);
    float h4b = lane_lo_bcast(hc[4], lane);
    v2f b2, b3;
    b2[0] = lo ? xc[CH2][0] : h0b;
    b2[1] = lo ? xc[CH2][1] : h1b;
    b3[0] = lo ? hc[2] : h4b;
    b3[1] = lo ? hc[3] : 0.0f;
    v2f bch[NCH];
#pragma unroll
    for (int c = 0; c < CH2; ++c) bch[c] = xc[c];
    bch[CH2] = b2;
    bch[CH3] = b3;

    // gates[gtype] = bias + Wcat * [x;h]^T      (gates on M, batch on N)
    v8f D[4];
#pragma unroll
    for (int g = 0; g < 4; ++g) {
      v8f d = bc[g];
#pragma unroll
      for (int c = 0; c < NCH; ++c) d = wmma4(a[g][c], bch[c], d);
      D[g] = d;
    }

    // Elementwise cell update. Valid data: elements 0..4, lanes 0..15
    // (i,f,g,o all sit at the same (element,lane) across the 4 accumulators).
#pragma unroll
    for (int e = 0; e < Hdim; ++e) {
      float iv = fast_sig(D[0][e]);
      float fv = fast_sig(D[1][e]);
      float gv = fast_tanh(D[2][e]);
      float ov = fast_sig(D[3][e]);
      cc[e] = fv * cc[e] + iv * gv;
      hc[e] = ov * fast_tanh(cc[e]);
    }

    if (lo) {
      float* o = dst + ((size_t)t * NB + bl) * 10 + dir * Hdim;
#pragma unroll
      for (int e = 0; e < Hdim; ++e) o[e] = hc[e];
    }

#pragma unroll
    for (int c = 0; c < NCH; ++c) xc[c] = xn[c];
  }
}

__global__ __launch_bounds__(64) void lstm_bidir_kernel(
    const float* __restrict__ x,     const float* __restrict__ W_ih0,
    const float* __restrict__ W_ih,  const float* __restrict__ W_hh,
    const float* __restrict__ bvec,  const float* __restrict__ W_fc,
    const float* __restrict__ b_fc,  float* __restrict__ out,
    float* __restrict__ buf0, float* __restrict__ buf1) {
  const int lane = threadIdx.x & 31;
  const int dir  = threadIdx.x >> 5;   // wave 0 = forward, wave 1 = backward
  const int b0   = blockIdx.x * BTILE;

  // ---- layer 0: x -> buf0 ------------------------------------------------
  {
    const float* Wi = W_ih0 + dir * (4 * Hdim * 2);
    const float* Wh = W_hh + (size_t)(0 * 2 + dir) * (4 * Hdim * Hdim);
    const float* bb = bvec + (size_t)(0 * 2 + dir) * (4 * Hdim);
    lstm_layer_dir<2>(x, buf0, Wi, Wh, bb, b0, lane, dir);
  }
  __threadfence_block();
  __syncthreads();

  // ---- layers 1..4: ping-pong buf0 <-> buf1 ------------------------------
  for (int l = 1; l < 5; ++l) {
    const float* src = (l & 1) ? buf0 : buf1;
    float*       dst = (l & 1) ? buf1 : buf0;
    const float* Wi = W_ih + (size_t)((l - 1) * 2 + dir) * (4 * Hdim * 2 * Hdim);
    const float* Wh = W_hh + (size_t)(l * 2 + dir) * (4 * Hdim * Hdim);
    const float* bb = bvec + (size_t)(l * 2 + dir) * (4 * Hdim);
    lstm_layer_dir<10>(src, dst, Wi, Wh, bb, b0, lane, dir);
    __threadfence_block();
    __syncthreads();
  }

  // ---- FC head on hs[T-1] (layer 4 output lives in buf0) -----------------
  if (dir == 0 && lane < BTILE) {
    const int b = b0 + lane;
    const float* hs = buf0 + ((size_t)(NT - 1) * NB + b) * 10;
    float acc0 = b_fc[0], acc1 = b_fc[1];
#pragma unroll
    for (int j = 0; j < 10; ++j) {
      float v = hs[j];
      acc0 += v * W_fc[j];       // W_fc[0][j]
      acc1 += v * W_fc[10 + j];  // W_fc[1][j]
    }
    out[b * 2 + 0] = acc0;
    out[b * 2 + 1] = acc1;
  }
}

extern "C" void kernel_launch(void* const* d_in, const int* in_sizes, int n_in,
                              void* d_out, int out_size, void* d_ws, size_t ws_size,
                              hipStream_t stream) {
  (void)in_sizes; (void)n_in; (void)out_size; (void)ws_size;
  const float* x     = (const float*)d_in[0];
  const float* W_ih0 = (const float*)d_in[1];
  const float* W_ih  = (const float*)d_in[2];
  const float* W_hh  = (const float*)d_in[3];
  const float* b     = (const float*)d_in[4];
  const float* W_fc  = (const float*)d_in[5];
  const float* b_fc  = (const float*)d_in[6];
  float* out = (float*)d_out;

  // Two ping-pong activation buffers: 2 * T*B*10 f32 = 83.9 MB (L2-resident)
  float* buf0 = (float*)d_ws;
  float* buf1 = buf0 + (size_t)NT * NB * 10;

  lstm_bidir_kernel<<<NB / BTILE, 64, 0, stream>>>(
      x, W_ih0, W_ih, W_hh, b, W_fc, b_fc, out, buf0, buf1);
}